// HadamardTransform_29781303230775
// MI455X (gfx1250) — compile-verified
//
#include <hip/hip_runtime.h>

// ---------------------------------------------------------------------------
// FWHT-128 on MI455X (gfx1250) via V_WMMA_F32_16X16X4_F32.
//
// H128 = H16 (x) H8.  One wave processes two groups (256 floats) as a 16x16
// matrix M = [X0|X1]:   Y = H16 * (M * blockdiag(H8,H8))
// -> 8 fp32 WMMA (16x16xK accumulated in K=4 chunks) per 256 elements.
// Memory-bound: 512 MiB @ 23.3 TB/s ~= 23 us floor; compute is free.
//
// WMMA register layouts used (per CDNA5 ISA, wave32):
//   A (16x4 chunk t, vgpr j): lane l holds A[l&15][4t + 2*(l>>4) + j]
//   B (4x16 chunk t, vgpr j): lane l holds B[4t + 2*(l>>4) + j][l&15]
//   C/D (16x16, vgpr r):      lane l holds D[r + 8*(l>>4)][l&15]
// C/D -> B conversion for an untransposed matrix is a pure lane-half swap:
// 8x ds_swizzle_b32 SWAPX16 (executed wave-wide, full EXEC) + 8x v_cndmask.
// ---------------------------------------------------------------------------

typedef __attribute__((ext_vector_type(2))) float v2f;
typedef __attribute__((ext_vector_type(8))) float v8f;

#define SWZ_SWAP16 0x401f  // group-of-32 swizzle: and=0x1f, or=0, xor=0x10

__device__ __forceinline__ float swap16(float x) {
    return __int_as_float(__builtin_amdgcn_ds_swizzle(__float_as_int(x), SWZ_SWAP16));
}

__global__ __launch_bounds__(256)
void fwht128_wmma_kernel(const float* __restrict__ x, float* __restrict__ out,
                         long long npairs, long long tail_group)
{
    const int lane = threadIdx.x & 31;
    const int hl   = lane >> 4;    // lane half (0: lanes 0-15, 1: lanes 16-31)
    const int m    = lane & 15;    // A-row / B-col / D-col index for this lane

    const long long wid     = (long long)blockIdx.x * (blockDim.x >> 5) + (threadIdx.x >> 5);
    const long long wstride = (long long)gridDim.x * (blockDim.x >> 5);

    const float scale = 0x1.6a09e6p-4f;  // 1/sqrt(128)

    // Constant operands, hoisted out of the streaming loop.
    //   a16[t][j] : A-layout H16 (scaled)     -> A operand of matmul #2
    //   b8 [t][j] : B-layout blockdiag(H8,H8) -> B operand of matmul #1
    float a16[4][2], b8[4][2];
#pragma unroll
    for (int t = 0; t < 4; ++t) {
#pragma unroll
        for (int j = 0; j < 2; ++j) {
            const int k = 4 * t + 2 * hl + j;
            a16[t][j] = (__builtin_popcount(m & k) & 1) ? -scale : scale;
            b8[t][j]  = ((k >> 3) == (m >> 3))
                          ? ((__builtin_popcount(k & m & 7) & 1) ? -1.0f : 1.0f)
                          : 0.0f;
        }
    }

    for (long long p = wid; p < npairs; p += wstride) {
        // ---- load M = [X0|X1] directly in WMMA A layout -------------------
        // lane needs M[m][k] = x[p*256 + 128*(k>>3) + 8*m + (k&7)],
        // k = 4t+2*hl+j -> per-lane base (8m + 2*hl) + imm {0,1,4,5,128,129,132,133}
        // (adjacent j pairs merge into global_load_b64)
        const float* __restrict__ px = x + p * 256 + (8 * m + 2 * hl);
        float md[4][2];
        md[0][0] = px[0];   md[0][1] = px[1];
        md[1][0] = px[4];   md[1][1] = px[5];
        md[2][0] = px[128]; md[2][1] = px[129];
        md[3][0] = px[132]; md[3][1] = px[133];

        // ---- matmul #1: W = M * blockdiag(H8,H8)  (K=16 as 4 chunks) ------
        v8f w = {};
#pragma unroll
        for (int t = 0; t < 4; ++t) {
            v2f a = {md[t][0], md[t][1]};
            v2f b = {b8[t][0], b8[t][1]};
            w = __builtin_amdgcn_wmma_f32_16x16x4_f32(false, a, false, b,
                                                      (short)0, w, false, false);
        }

        // ---- C/D layout -> B layout (lane-half swap only) -----------------
        // Swizzles MUST run wave-wide (full EXEC): hoist them out of the
        // selects so they are unconditional straight-line code, then cndmask.
        float sw[8];
#pragma unroll
        for (int r = 0; r < 8; ++r) sw[r] = swap16(w[r]);

        float wb[4][2];
#pragma unroll
        for (int j = 0; j < 2; ++j) {
            wb[0][j] = hl ? sw[2 + j] : w[j];
            wb[1][j] = hl ? sw[6 + j] : w[4 + j];
            wb[2][j] = hl ? w[2 + j]  : sw[j];
            wb[3][j] = hl ? w[6 + j]  : sw[4 + j];
        }

        // ---- matmul #2: Y = (scale*H16) * W -------------------------------
        v8f y = {};
#pragma unroll
        for (int t = 0; t < 4; ++t) {
            v2f a = {a16[t][0], a16[t][1]};
            v2f b = {wb[t][0], wb[t][1]};
            y = __builtin_amdgcn_wmma_f32_16x16x4_f32(false, a, false, b,
                                                      (short)0, y, false, false);
        }

        // ---- store Y from C/D layout --------------------------------------
        // lane l, vgpr r -> out[p*256 + 128*(m>>3) + (m&7) + 64*hl + 8*r]
        float* __restrict__ po = out + p * 256 + (128 * (m >> 3) + (m & 7) + 64 * hl);
#pragma unroll
        for (int r = 0; r < 8; ++r) po[8 * r] = y[r];
    }

    // ---- odd trailing group (not hit for the reference shape) -------------
    if (tail_group >= 0 && wid == 0) {
        const float* gx = x + tail_group * 128;
        float* go = out + tail_group * 128;
        // index i = lane + 32*s ; FWHT stages commute across bits.
        float v0 = gx[lane], v1 = gx[lane + 32], v2 = gx[lane + 64], v3 = gx[lane + 96];
#define FWHT_XSTAGE(MASK)                                                          \
        do {                                                                       \
            const int off = ((MASK) << 10) | 0x1f;                                 \
            float o0 = __int_as_float(__builtin_amdgcn_ds_swizzle(__float_as_int(v0), off)); \
            float o1 = __int_as_float(__builtin_amdgcn_ds_swizzle(__float_as_int(v1), off)); \
            float o2 = __int_as_float(__builtin_amdgcn_ds_swizzle(__float_as_int(v2), off)); \
            float o3 = __int_as_float(__builtin_amdgcn_ds_swizzle(__float_as_int(v3), off)); \
            const bool up = (lane & (MASK)) != 0;                                  \
            v0 = up ? (o0 - v0) : (v0 + o0);                                       \
            v1 = up ? (o1 - v1) : (v1 + o1);                                       \
            v2 = up ? (o2 - v2) : (v2 + o2);                                       \
            v3 = up ? (o3 - v3) : (v3 + o3);                                       \
        } while (0)
        FWHT_XSTAGE(1); FWHT_XSTAGE(2); FWHT_XSTAGE(4); FWHT_XSTAGE(8); FWHT_XSTAGE(16);
#undef FWHT_XSTAGE
        // bits 5/6 live in s (register index)
        float t0 = v0 + v1, t1 = v0 - v1, t2 = v2 + v3, t3 = v2 - v3;
        v0 = t0 + t2; v1 = t1 + t3; v2 = t0 - t2; v3 = t1 - t3;
        go[lane]      = v0 * scale;
        go[lane + 32] = v1 * scale;
        go[lane + 64] = v2 * scale;
        go[lane + 96] = v3 * scale;
    }
}

extern "C" void kernel_launch(void* const* d_in, const int* in_sizes, int n_in,
                              void* d_out, int out_size, void* d_ws, size_t ws_size,
                              hipStream_t stream) {
    const float* x = (const float*)d_in[0];
    float* out = (float*)d_out;

    const long long n       = (long long)in_sizes[0];  // total fp32 elements
    const long long ngroups = n / 128;
    const long long npairs  = ngroups / 2;
    const long long tail    = (ngroups & 1) ? (ngroups - 1) : -1;

    const int wavesPerBlock = 8;  // 256 threads = 8 wave32
    long long blocksNeeded = (npairs + wavesPerBlock - 1) / wavesPerBlock;
    if (blocksNeeded < 1) blocksNeeded = 1;
    if (blocksNeeded > 4096) blocksNeeded = 4096;  // persistent grid-stride

    fwht128_wmma_kernel<<<(int)blocksNeeded, 256, 0, stream>>>(x, out, npairs, tail);
}